// Parameter_Estimator_18047452577915
// MI455X (gfx1250) — compile-verified
//
#include <hip/hip_runtime.h>
#include <hip/hip_bf16.h>
#include <math.h>

// Adam hyperparameters from the reference
#define ADAM_B1 0.9f
#define ADAM_B2 0.999f
#define ADAM_EPS 1e-8f
#define ADAM_LR 0.02f

__device__ __forceinline__ float frcp(float x)  { return __builtin_amdgcn_rcpf(x); }
__device__ __forceinline__ float fsqrt(float x) { return __builtin_amdgcn_sqrtf(x); }

__global__ __launch_bounds__(128) void param_est_adam_kernel(
    const float* __restrict__ X,        // (B,5,5)
    const float* __restrict__ Y,        // (B,5)
    const float* __restrict__ ch0p,     // (B,1,3)
    const float* __restrict__ pf,       // (14,)   uniform
    const float* __restrict__ x_a,      // (3,)    uniform
    const float* __restrict__ s_a_inv,  // (3,3)   uniform
    const float* __restrict__ s_e_inv,  // (5,5)   uniform
    const int*   __restrict__ nitp,     // (1,)    uniform
    float* __restrict__ out,            // (B,9)
    int B)
{
    const int b = blockIdx.x * blockDim.x + threadIdx.x;
    if (b >= B) return;

    const float* __restrict__ Xe = X + (size_t)b * 25;
    const float* __restrict__ Ye = Y + (size_t)b * 5;
    const float* __restrict__ Ce = ch0p + (size_t)b * 3;

    // CDNA5 prefetch (global_prefetch_b8): per-lane rows in flight while the
    // wave evaluates the uniform SGPR constants below.
    __builtin_prefetch(Xe, 0, 3);
    __builtin_prefetch(Ye, 0, 3);
    __builtin_prefetch(Ce, 0, 3);

    const int nit = nitp[0];

    // Normalized wavelengths: (lambda - 440)/100
    const float lam[5] = { -0.275f, 0.025f, 0.50f, 0.70f, 1.15f };

    // ---- wave-uniform precompute (scalar-promoted: SGPRs + s_*_f32 SALU) ----
    float pfv[14];
    #pragma unroll
    for (int i = 0; i < 14; ++i) pfv[i] = pf[i];

    float aphc[5], acdc[5], bbcc[5];
    #pragma unroll
    for (int i = 0; i < 5; ++i) {
        aphc[i] = pfv[0] * 0.05f * (1.0f + pfv[1] * 0.1f * lam[i] + pfv[2] * 0.01f);
        acdc[i] = pfv[5] * 0.1f * __expf(-pfv[6] * 1.7f * lam[i]);
        bbcc[i] = pfv[3] * 0.001f * (1.0f + pfv[4] * 0.05f * lam[i]);
    }
    const float anapc  = pfv[12] * 0.005f;   // coef of nap in a
    const float bnapc  = pfv[13] * 0.005f;   // coef of nap in bb
    const float rA     = pfv[7] * 0.089f;
    const float rB     = pfv[8] * 0.1245f;
    const float rB2    = 2.0f * rB;
    const float kf     = pfv[9] * 0.9f + pfv[10] * 0.1f;
    const float bscale = pfv[11];

    // Symmetrized inverse covariances: grad of 0.5*r^T S r is 0.5*(S+S^T) r
    float Me[5][5];
    #pragma unroll
    for (int i = 0; i < 5; ++i)
        #pragma unroll
        for (int j = 0; j < 5; ++j)
            Me[i][j] = 0.5f * (s_e_inv[i * 5 + j] + s_e_inv[j * 5 + i]);
    float Ma[3][3];
    #pragma unroll
    for (int i = 0; i < 3; ++i)
        #pragma unroll
        for (int j = 0; j < 3; ++j)
            Ma[i][j] = 0.5f * (s_a_inv[i * 3 + j] + s_a_inv[j * 3 + i]);
    const float xa0 = x_a[0], xa1 = x_a[1], xa2 = x_a[2];

    // ---- per-element loads ----
    float f[5], x3[5], y[5];
    #pragma unroll
    for (int i = 0; i < 5; ++i) {
        const float Edir = Xe[i * 5 + 0];
        const float Edif = Xe[i * 5 + 1];
        x3[i] = Xe[i * 5 + 3];
        f[i]  = (Edir + 0.5f * Edif) * frcp(Edir + Edif);
        y[i]  = Ye[i];
    }
    float ch[3] = { Ce[0], Ce[1], Ce[2] };
    float m[3]  = { 0.f, 0.f, 0.f };
    float v[3]  = { 0.f, 0.f, 0.f };
    float p1 = 1.0f, p2 = 1.0f;   // B1^t, B2^t

    // ---- Adam loop with analytic Jacobian ----
    for (int t = 0; t < nit; ++t) {
        const float chla = __expf(ch[0]);
        const float nap  = __expf(ch[1]);
        const float cdom = __expf(ch[2]);

        // d(a)/dch1 and d(bb)/dch1 (wavelength-independent)
        const float da1 = anapc * nap;
        const float db1 = bnapc * nap;

        float r[5], J0[5], J1[5], J2[5];
        #pragma unroll
        for (int i = 0; i < 5; ++i) {
            const float a   = fmaf(anapc, nap, fmaf(acdc[i], cdom, fmaf(aphc[i], chla, 0.0045f)));
            const float bb  = fmaf(bnapc, nap, fmaf(bbcc[i], chla, 0.0012f));
            const float inv = frcp(a + bb);
            const float u   = bb * inv;
            const float tu  = fmaf(rB, u, rA) * u;
            r[i] = fmaf(tu, f[i], -y[i]);
            const float drdu = fmaf(rB2, u, rA) * f[i];
            const float k    = (drdu * inv) * inv;
            // J = k * (dbb*a - bb*da), factored per parameter:
            J0[i] = (k * chla) * fmaf(bbcc[i], a, -(aphc[i] * bb));
            J1[i] = k * fmaf(db1, a, -(da1 * bb));
            J2[i] = -(k * cdom) * (acdc[i] * bb);
        }

        const float d0 = ch[0] - xa0, d1 = ch[1] - xa1, d2 = ch[2] - xa2;
        float g0 = Ma[0][0] * d0 + Ma[0][1] * d1 + Ma[0][2] * d2;
        float g1 = Ma[1][0] * d0 + Ma[1][1] * d1 + Ma[1][2] * d2;
        float g2 = Ma[2][0] * d0 + Ma[2][1] * d1 + Ma[2][2] * d2;
        #pragma unroll
        for (int i = 0; i < 5; ++i) {
            float Ser = 0.0f;
            #pragma unroll
            for (int j = 0; j < 5; ++j) Ser = fmaf(Me[i][j], r[j], Ser);
            g0 = fmaf(J0[i], Ser, g0);
            g1 = fmaf(J1[i], Ser, g1);
            g2 = fmaf(J2[i], Ser, g2);
        }
        const float g[3] = { g0, g1, g2 };

        // Adam: LR*mh/(sqrt(vh)+eps) = (LR*c1)*m / (sqrt(c2)*sqrt(v) + eps)
        p1 *= ADAM_B1;
        p2 *= ADAM_B2;
        const float lr1 = ADAM_LR * frcp(1.0f - p1);   // LR / (1 - B1^t)
        const float s2  = fsqrt(frcp(1.0f - p2));      // 1/sqrt(1 - B2^t)
        #pragma unroll
        for (int k = 0; k < 3; ++k) {
            m[k] = fmaf(ADAM_B1, m[k], (1.0f - ADAM_B1) * g[k]);
            v[k] = fmaf(ADAM_B2, v[k], (1.0f - ADAM_B2) * (g[k] * g[k]));
            const float den = fmaf(s2, fsqrt(v[k]), ADAM_EPS);
            ch[k] = fmaf(-(lr1 * m[k]), frcp(den), ch[k]);
        }
    }

    // ---- outputs: [ch0, kd(5), bbp(3)] ----
    // Write-once streaming data: non-temporal stores (th:TH_STORE_NT) keep the
    // 3.6 MB output from dirtying L2 lines nothing will re-read.
    const float chla = __expf(ch[0]);
    const float nap  = __expf(ch[1]);
    const float cdom = __expf(ch[2]);
    float* __restrict__ ob = out + (size_t)b * 9;
    __builtin_nontemporal_store(ch[0], &ob[0]);
    #pragma unroll
    for (int i = 0; i < 5; ++i) {
        const float a   = fmaf(anapc, nap, fmaf(acdc[i], cdom, fmaf(aphc[i], chla, 0.0045f)));
        const float bb  = fmaf(bnapc, nap, fmaf(bbcc[i], chla, 0.0012f));
        const float sig = frcp(1.0f + __expf(-x3[i]));
        const float mu  = 0.5f + 0.5f * sig;
        __builtin_nontemporal_store(((a + bb) * frcp(mu)) * kf, &ob[1 + i]);
    }
    // bbp wavelengths {1, 2, 4}
    const float bnap = bnapc * nap;
    __builtin_nontemporal_store(bscale * fmaf(bbcc[1], chla, bnap), &ob[6]);
    __builtin_nontemporal_store(bscale * fmaf(bbcc[2], chla, bnap), &ob[7]);
    __builtin_nontemporal_store(bscale * fmaf(bbcc[4], chla, bnap), &ob[8]);
}

extern "C" void kernel_launch(void* const* d_in, const int* in_sizes, int n_in,
                              void* d_out, int out_size, void* d_ws, size_t ws_size,
                              hipStream_t stream) {
    (void)n_in; (void)out_size; (void)d_ws; (void)ws_size;
    const float* X    = (const float*)d_in[0];
    const float* Y    = (const float*)d_in[1];
    const float* ch0  = (const float*)d_in[2];
    const float* pf   = (const float*)d_in[3];
    const float* xa   = (const float*)d_in[4];
    const float* sa   = (const float*)d_in[5];
    const float* se   = (const float*)d_in[6];
    const int*   nit  = (const int*)d_in[7];
    float* out        = (float*)d_out;

    const int B = in_sizes[1] / 5;   // Y is (B,5)
    const int block = 128;           // 4 wave32s per workgroup, finer WGP spread
    const int grid  = (B + block - 1) / block;
    param_est_adam_kernel<<<grid, block, 0, stream>>>(X, Y, ch0, pf, xa, sa, se, nit, out, B);
}